// OT_Attn_assem_61795989455392
// MI455X (gfx1250) — compile-verified
//
#include <hip/hip_runtime.h>
#include <hip/hip_bf16.h>
#include <math.h>

typedef float v2f __attribute__((ext_vector_type(2)));
typedef float v8f __attribute__((ext_vector_type(8)));

#define N_X 4096
#define M_Y 2048
#define DIM 1024
#define OT_REG 0.1f
#define FI_EXP 0.83333331f   // 0.5/(0.5+0.1)
#define N_ITERS 100

// ---------------------------------------------------------------------------
// Kernel 0: init v = 1/M, zero max-accumulator and dist accumulator
// ---------------------------------------------------------------------------
__global__ void init_kernel(float* __restrict__ v, unsigned int* __restrict__ maxBits,
                            float* __restrict__ distOut) {
    int i = blockIdx.x * blockDim.x + threadIdx.x;
    if (i < M_Y) v[i] = 1.0f / (float)M_Y;
    if (i == 0) { *maxBits = 0u; *distOut = 0.0f; }
}

// ---------------------------------------------------------------------------
// Kernel 1: per-row min-shift + squared norm. One wave32 per row.
// Rows [0, N_X) come from x, rows [N_X, N_X+M_Y) from y.
// ---------------------------------------------------------------------------
__global__ __launch_bounds__(256)
void prep_rows(const float* __restrict__ x, const float* __restrict__ y,
               float* __restrict__ xs, float* __restrict__ ys,
               float* __restrict__ xn, float* __restrict__ yn) {
    int wave = (blockIdx.x * blockDim.x + threadIdx.x) >> 5;
    int lane = threadIdx.x & 31;
    const float* src; float* dst; float* nrm; int row;
    if (wave < N_X) { src = x; dst = xs; nrm = xn; row = wave; }
    else {
        row = wave - N_X;
        if (row >= M_Y) return;
        src = y; dst = ys; nrm = yn;
    }
    const float4* p = (const float4*)(src + (size_t)row * DIM);
    float4 vals[8];
    float mn = 3.4e38f;
#pragma unroll
    for (int it = 0; it < 8; ++it) {
        vals[it] = p[it * 32 + lane];
        mn = fminf(mn, fminf(fminf(vals[it].x, vals[it].y), fminf(vals[it].z, vals[it].w)));
    }
#pragma unroll
    for (int s = 16; s > 0; s >>= 1) mn = fminf(mn, __shfl_xor(mn, s, 32));
    float sq = 0.0f;
    float4* q = (float4*)(dst + (size_t)row * DIM);
#pragma unroll
    for (int it = 0; it < 8; ++it) {
        float4 t = vals[it];
        t.x -= mn; t.y -= mn; t.z -= mn; t.w -= mn;
        sq += t.x * t.x + t.y * t.y + t.z * t.z + t.w * t.w;
        q[it * 32 + lane] = t;
    }
#pragma unroll
    for (int s = 16; s > 0; s >>= 1) sq += __shfl_xor(sq, s, 32);
    if (lane == 0) nrm[row] = sq;
}

// ---------------------------------------------------------------------------
// Kernel 2: C = clamp(xn + yn - 2*xs@ys^T, 0) via V_WMMA_F32_16X16X4_F32.
// Block tile 128x128, 8 waves, each wave a 64x32 region (4x2 16x16 tiles).
// Fused block max(C) -> uint-bits atomic max (valid: C >= 0).
// ---------------------------------------------------------------------------
#define BM 128
#define BN 128
#define KC 32
#define LDT 36   // padded row (floats): 16B aligned rows, conflict-light b64 reads

__global__ __launch_bounds__(256)
void cost_gemm(const float* __restrict__ xs, const float* __restrict__ ys,
               const float* __restrict__ xn, const float* __restrict__ yn,
               float* __restrict__ C, unsigned int* __restrict__ maxBits) {
    __shared__ float lA[BM * LDT];
    __shared__ float lB[BN * LDT];
    const int tid  = threadIdx.x;
    const int lane = tid & 31;
    const int wave = tid >> 5;
    const int wm = wave >> 2;      // 0..1  -> 64-row slab
    const int wn = wave & 3;       // 0..3  -> 32-col slab
    const int bm0 = blockIdx.y * BM;
    const int bn0 = blockIdx.x * BN;
    const int half = lane >> 4;    // K half selector for f32 A/B fragments
    const int l16  = lane & 15;

    v8f acc[4][2];
#pragma unroll
    for (int mt = 0; mt < 4; ++mt)
#pragma unroll
        for (int nt = 0; nt < 2; ++nt)
            acc[mt][nt] = (v8f){0.f, 0.f, 0.f, 0.f, 0.f, 0.f, 0.f, 0.f};

    const int lrow = tid >> 3;        // 0..31
    const int lcol = (tid & 7) * 4;   // float4 column

    for (int kc = 0; kc < DIM; kc += KC) {
#pragma unroll
        for (int ps = 0; ps < 4; ++ps) {
            int r = ps * 32 + lrow;
            float4 ta = *(const float4*)(xs + (size_t)(bm0 + r) * DIM + kc + lcol);
            *(float4*)(lA + r * LDT + lcol) = ta;
            float4 tb = *(const float4*)(ys + (size_t)(bn0 + r) * DIM + kc + lcol);
            *(float4*)(lB + r * LDT + lcol) = tb;
        }
        __syncthreads();
#pragma unroll
        for (int k = 0; k < KC; k += 4) {
            v2f af[4], bf[2];
#pragma unroll
            for (int mt = 0; mt < 4; ++mt)
                af[mt] = *(const v2f*)(lA + (wm * 64 + mt * 16 + l16) * LDT + k + half * 2);
#pragma unroll
            for (int nt = 0; nt < 2; ++nt)
                bf[nt] = *(const v2f*)(lB + (wn * 32 + nt * 16 + l16) * LDT + k + half * 2);
#pragma unroll
            for (int mt = 0; mt < 4; ++mt)
#pragma unroll
                for (int nt = 0; nt < 2; ++nt)
                    acc[mt][nt] = __builtin_amdgcn_wmma_f32_16x16x4_f32(
                        false, af[mt], false, bf[nt],
                        (short)0, acc[mt][nt], false, false);
        }
        __syncthreads();
    }

    // Epilogue: C = clamp(xn + yn - 2*dot, 0); track block max.
    float localMax = 0.0f;
#pragma unroll
    for (int mt = 0; mt < 4; ++mt) {
#pragma unroll
        for (int nt = 0; nt < 2; ++nt) {
            int gn = bn0 + wn * 32 + nt * 16 + l16;
            float ynv = yn[gn];
#pragma unroll
            for (int e = 0; e < 8; ++e) {
                int gm = bm0 + wm * 64 + mt * 16 + half * 8 + e;
                float val = xn[gm] + ynv - 2.0f * acc[mt][nt][e];
                val = fmaxf(val, 0.0f);
                C[(size_t)gm * M_Y + gn] = val;
                localMax = fmaxf(localMax, val);
            }
        }
    }
#pragma unroll
    for (int s = 16; s > 0; s >>= 1) localMax = fmaxf(localMax, __shfl_xor(localMax, s, 32));
    if (lane == 0) atomicMax(maxBits, __float_as_uint(localMax));
}

// ---------------------------------------------------------------------------
// Kernel 3: K = exp(-C / (reg * maxC)), plus KT via LDS-tiled transpose.
// ---------------------------------------------------------------------------
__global__ __launch_bounds__(256)
void expk_kernel(const float* __restrict__ C, const unsigned int* __restrict__ maxBits,
                 float* __restrict__ K, float* __restrict__ KT) {
    __shared__ float tile[32][33];
    float invS = 1.0f / (OT_REG * __uint_as_float(*maxBits));
    int j0 = blockIdx.x * 32, i0 = blockIdx.y * 32;
    int tx = threadIdx.x, ty = threadIdx.y;   // block (32, 8)
#pragma unroll
    for (int r = 0; r < 4; ++r) {
        int i = i0 + ty + r * 8;
        size_t idx = (size_t)i * M_Y + j0 + tx;
        float k = __expf(-C[idx] * invS);
        K[idx] = k;
        tile[ty + r * 8][tx] = k;
    }
    __syncthreads();
#pragma unroll
    for (int r = 0; r < 4; ++r) {
        int j = j0 + ty + r * 8;
        KT[(size_t)j * N_X + i0 + tx] = tile[tx][ty + r * 8];
    }
}

// ---------------------------------------------------------------------------
// Kernel 4: one Sinkhorn half-step. One wave32 per row of `mat`:
//   out[r] = (marg / dot(mat[r], in))^fi
// ---------------------------------------------------------------------------
__global__ __launch_bounds__(256)
void sink_step(const float* __restrict__ mat, const float* __restrict__ in,
               float* __restrict__ out, int rows, int cols, float marg) {
    int wave = (blockIdx.x * blockDim.x + threadIdx.x) >> 5;
    int lane = threadIdx.x & 31;
    if (wave >= rows) return;
    const float4* kp = (const float4*)(mat + (size_t)wave * cols);
    const float4* vp = (const float4*)in;
    float dot = 0.0f;
    int iters = cols >> 7;   // 128 floats per wave-iteration
    for (int it = 0; it < iters; ++it) {
        float4 kk = kp[it * 32 + lane];
        float4 vv = vp[it * 32 + lane];
        dot += kk.x * vv.x + kk.y * vv.y + kk.z * vv.z + kk.w * vv.w;
    }
#pragma unroll
    for (int s = 16; s > 0; s >>= 1) dot += __shfl_xor(dot, s, 32);
    if (lane == 0) out[wave] = __powf(marg / dot, FI_EXP);
}

// ---------------------------------------------------------------------------
// Kernel 5: out[j, i] = u[i] * KT[j, i] * v[j]   (flow transposed, [M, N])
// ---------------------------------------------------------------------------
__global__ __launch_bounds__(256)
void flow_kernel(const float* __restrict__ KT, const float* __restrict__ u,
                 const float* __restrict__ v, float* __restrict__ out) {
    int j = blockIdx.y;
    int i = blockIdx.x * blockDim.x + threadIdx.x;
    size_t idx = (size_t)j * N_X + i;
    out[idx] = u[i] * KT[idx] * v[j];
}

// ---------------------------------------------------------------------------
// Kernel 6: dist = sum_ij C_ij * u_i * K_ij * v_j  (grid-stride + atomicAdd)
// ---------------------------------------------------------------------------
__global__ __launch_bounds__(256)
void dist_kernel(const float* __restrict__ C, const float* __restrict__ K,
                 const float* __restrict__ u, const float* __restrict__ v,
                 float* __restrict__ dist) {
    __shared__ float wsum[8];
    size_t idx = (size_t)blockIdx.x * blockDim.x + threadIdx.x;
    size_t stride = (size_t)gridDim.x * blockDim.x;
    const size_t total = (size_t)N_X * M_Y;
    float s = 0.0f;
    for (size_t p = idx; p < total; p += stride) {
        int i = (int)(p >> 11);          // / M_Y (2048)
        int j = (int)(p & (M_Y - 1));
        s += C[p] * u[i] * K[p] * v[j];
    }
#pragma unroll
    for (int d = 16; d > 0; d >>= 1) s += __shfl_xor(s, d, 32);
    int lane = threadIdx.x & 31, wave = threadIdx.x >> 5;
    if (lane == 0) wsum[wave] = s;
    __syncthreads();
    if (wave == 0) {
        float t = (lane < 8) ? wsum[lane] : 0.0f;
#pragma unroll
        for (int d = 4; d > 0; d >>= 1) t += __shfl_xor(t, d, 32);
        if (lane == 0) atomicAdd(dist, t);
    }
}

// ---------------------------------------------------------------------------
extern "C" void kernel_launch(void* const* d_in, const int* in_sizes, int n_in,
                              void* d_out, int out_size, void* d_ws, size_t ws_size,
                              hipStream_t stream) {
    const float* x = (const float*)d_in[0];   // [1, 4096, 1024]
    const float* y = (const float*)d_in[1];   // [1, 2048, 1024]
    float* out = (float*)d_out;               // [1, M, N] flow.T then dist scalar

    // Workspace carving (~126 MB total, L2-resident hot set)
    float* ws  = (float*)d_ws;
    float* xs  = ws;                                    // 4096*1024
    float* ys  = xs + (size_t)N_X * DIM;                // 2048*1024
    float* xn  = ys + (size_t)M_Y * DIM;                // 4096
    float* yn  = xn + N_X;                              // 2048
    float* Cm  = yn + M_Y;                              // 4096*2048
    float* Km  = Cm + (size_t)N_X * M_Y;                // 4096*2048
    float* KTm = Km + (size_t)N_X * M_Y;                // 4096*2048
    float* u   = KTm + (size_t)N_X * M_Y;               // 4096
    float* v   = u + N_X;                               // 2048
    unsigned int* maxBits = (unsigned int*)(v + M_Y);   // 1

    float* distOut = out + (size_t)N_X * M_Y;

    // 0) init
    init_kernel<<<(M_Y + 255) / 256, 256, 0, stream>>>(v, maxBits, distOut);

    // 1) row prep: (4096 + 2048) waves, 8 waves per block
    prep_rows<<<(N_X + M_Y) / 8, 256, 0, stream>>>(x, y, xs, ys, xn, yn);

    // 2) WMMA cost GEMM + max reduction
    dim3 ggrid(M_Y / BN, N_X / BM);
    cost_gemm<<<ggrid, 256, 0, stream>>>(xs, ys, xn, yn, Cm, maxBits);

    // 3) Gibbs kernel + transpose
    dim3 egrid(M_Y / 32, N_X / 32);
    expk_kernel<<<egrid, dim3(32, 8), 0, stream>>>(Cm, maxBits, Km, KTm);

    // 4) 100 Sinkhorn iterations (both matrices stay in L2)
    for (int it = 0; it < N_ITERS; ++it) {
        sink_step<<<N_X / 8, 256, 0, stream>>>(Km, v, u, N_X, M_Y, 1.0f / (float)N_X);
        sink_step<<<M_Y / 8, 256, 0, stream>>>(KTm, u, v, M_Y, N_X, 1.0f / (float)M_Y);
    }

    // 5) flow.T output
    flow_kernel<<<dim3(N_X / 256, M_Y), 256, 0, stream>>>(KTm, u, v, out);

    // 6) OT distance
    dist_kernel<<<2048, 256, 0, stream>>>(Cm, Km, u, v, distOut);
}